// CentroidDistance_89919435309056
// MI455X (gfx1250) — compile-verified
//
#include <hip/hip_runtime.h>
#include <math.h>

#define NCENT 512
#define DIM   128
#define EPSF  1e-6f

typedef __attribute__((ext_vector_type(2))) float v2f;
typedef __attribute__((ext_vector_type(4))) float v4f;
typedef __attribute__((ext_vector_type(8))) float v8f;

// LDS pitch in floats: even (8B-aligned v2f) and 132 % 64banks = 4 -> 16 rows/cols
// map to 16 distinct banks for the WMMA fragment reads.
#define PITCH 132

// ---------------- K0: zero ws accumulators (colsum[512] + mask_sum[1]) ---------
__global__ void init_ws_k(float* wsf) {
    int i = blockIdx.x * blockDim.x + threadIdx.x;
    if (i < 768) wsf[i] = 0.0f;
}

// ---------------- K1: sum(mask) ------------------------------------------------
__global__ void mask_sum_k(const float* __restrict__ mask, int n, float* __restrict__ out) {
    float s = 0.0f;
    for (int i = blockIdx.x * blockDim.x + threadIdx.x; i < n; i += gridDim.x * blockDim.x)
        s += mask[i];
    #pragma unroll
    for (int o = 16; o >= 1; o >>= 1) s += __shfl_xor(s, o, 32);
    if ((threadIdx.x & 31) == 0) atomicAdd(out, s);
}

// ---------------- K2: centroids = exp_map_zero(emb), sign folded into dim 0 ----
// out[c][0]   = -sqrt(1 + s^2 * ldv),  s = sinh(min(nd,1)) / nd, nd = sqrt(max(ldv+eps,1e-12))
// out[c][d>0] =  s * emb[c][d]
__global__ void centroid_k(const float* __restrict__ emb, float* __restrict__ cent) {
    const int lane = threadIdx.x & 31;
    const int c = blockIdx.x * 8 + (threadIdx.x >> 5);
    if (c >= NCENT) return;
    const float* e = emb + (size_t)c * DIM;
    float v0 = (lane == 0) ? 0.0f : e[lane];
    float v1 = e[lane + 32], v2 = e[lane + 64], v3 = e[lane + 96];
    float ldv = v0 * v0 + v1 * v1 + v2 * v2 + v3 * v3;
    #pragma unroll
    for (int o = 16; o >= 1; o >>= 1) ldv += __shfl_xor(ldv, o, 32);
    float nd = sqrtf(fmaxf(ldv + EPSF, 1e-12f));
    float t  = fminf(nd, 1.0f);
    float s  = sinhf(t) / nd;
    float first = sqrtf(1.0f + s * s * ldv);
    float* o = cent + (size_t)c * DIM;
    o[lane]      = (lane == 0) ? -first : s * v0;
    o[lane + 32] = s * v1;
    o[lane + 64] = s * v2;
    o[lane + 96] = s * v3;
}

// ---------------- K3: GEMM (WMMA f32 16x16x4) + acosh + column sums ------------
// block = 256 thr = 8 waves; tile = 64 rows x 32 cols; wave (rowsub = w&3, colsub = w>>2)
// owns a 16x16 D tile. N is a multiple of 16, so row-subtiles are all-or-nothing valid.
__global__ __launch_bounds__(256, 2)
void dist_k(const float* __restrict__ node, const float* __restrict__ cent,
            float* __restrict__ out_node, float* __restrict__ colsum, int N) {
    __shared__ float sA[64 * PITCH];
    __shared__ float sB[32 * PITCH];
    __shared__ float sCol[32];

    const int tid  = threadIdx.x;
    const int lane = tid & 31;
    const int wave = tid >> 5;
    const int m0 = blockIdx.x * 64;
    const int c0 = blockIdx.y * 32;

    // Stage A tile (64 x 128): 2048 float4, 8 per thread; zero-pad rows >= N.
    for (int i = tid; i < 2048; i += 256) {
        int r = i >> 5, q = (i & 31) * 4;
        int grow = m0 + r;
        v4f val = {0.f, 0.f, 0.f, 0.f};
        if (grow < N)
            val = __builtin_nontemporal_load((const v4f*)(node + (size_t)grow * DIM + q));
        v2f lo = {val.x, val.y}, hi = {val.z, val.w};
        *(v2f*)(sA + r * PITCH + q)     = lo;
        *(v2f*)(sA + r * PITCH + q + 2) = hi;
    }
    // Stage B tile (32 x 128): 1024 float4, 4 per thread (centroids stay L2-hot).
    for (int i = tid; i < 1024; i += 256) {
        int r = i >> 5, q = (i & 31) * 4;
        v4f val = *(const v4f*)(cent + (size_t)(c0 + r) * DIM + q);
        v2f lo = {val.x, val.y}, hi = {val.z, val.w};
        *(v2f*)(sB + r * PITCH + q)     = lo;
        *(v2f*)(sB + r * PITCH + q + 2) = hi;
    }
    if (tid < 32) sCol[tid] = 0.0f;
    __syncthreads();

    const int rowsub = wave & 3, colsub = wave >> 2;
    const int rowStart = m0 + rowsub * 16;
    const bool valid = rowStart < N;   // N % 16 == 0 -> whole subtile valid or not

    v8f acc = {0.f, 0.f, 0.f, 0.f, 0.f, 0.f, 0.f, 0.f};
    const int sel = (lane < 16) ? 0 : 2;            // K-pair select per ISA A/B layout
    const int rl  = rowsub * 16 + (lane & 15);      // A: lane -> M row
    const int cl  = colsub * 16 + (lane & 15);      // B: lane -> N col

    if (valid) {
#if __has_builtin(__builtin_amdgcn_wmma_f32_16x16x4_f32)
        #pragma unroll
        for (int k0 = 0; k0 < DIM; k0 += 4) {
            v2f a = *(const v2f*)(sA + rl * PITCH + k0 + sel);
            v2f b = *(const v2f*)(sB + cl * PITCH + k0 + sel);
            acc = __builtin_amdgcn_wmma_f32_16x16x4_f32(false, a, false, b,
                                                        (short)0, acc, false, false);
        }
#else
        // Fallback: same accumulator layout via scalar FMA from LDS.
        const int fcol = colsub * 16 + (lane & 15);
        #pragma unroll
        for (int r = 0; r < 8; ++r) {
            int rloc = rowsub * 16 + r + ((lane < 16) ? 0 : 8);
            float s = 0.0f;
            for (int k = 0; k < DIM; ++k) s = fmaf(sA[rloc * PITCH + k], sB[fcol * PITCH + k], s);
            acc[r] = s;
        }
#endif
    }

    // Epilogue: dist = acosh(max(-ldot, 1+eps)); streaming store; column partials.
    float csum = 0.0f;
    if (valid) {
        const int col   = c0 + colsub * 16 + (lane & 15);
        const int rbase = rowStart + ((lane < 16) ? 0 : 8);
        #pragma unroll
        for (int r = 0; r < 8; ++r) {
            float x = fmaxf(-acc[r], 1.0f + EPSF);
            float d = logf(x + sqrtf(fmaf(x, x, -1.0f)));
            __builtin_nontemporal_store(d, out_node + (size_t)(rbase + r) * NCENT + col);
            csum += d;
        }
        csum += __shfl_xor(csum, 16, 32);           // fold the two row-halves per column
        if (lane < 16) atomicAdd(&sCol[colsub * 16 + lane], csum);   // ds_add_f32
    }
    __syncthreads();
    if (tid < 32) atomicAdd(&colsum[c0 + tid], sCol[tid]);           // global_atomic_add_f32
}

// ---------------- K4: graph[c] = colsum[c] / mask_sum --------------------------
__global__ void finalize_k(const float* __restrict__ colsum, const float* __restrict__ msum,
                           float* __restrict__ out_graph) {
    int c = blockIdx.x * blockDim.x + threadIdx.x;
    if (c < NCENT) out_graph[c] = colsum[c] / msum[0];
}

extern "C" void kernel_launch(void* const* d_in, const int* in_sizes, int n_in,
                              void* d_out, int out_size, void* d_ws, size_t ws_size,
                              hipStream_t stream) {
    const float* node = (const float*)d_in[0];   // (N, 128)
    const float* mask = (const float*)d_in[1];   // (N, 1)
    const float* emb  = (const float*)d_in[2];   // (512, 128)
    const int N      = in_sizes[0] / DIM;
    const int nMask  = in_sizes[1];

    float* wsf     = (float*)d_ws;
    float* colsum  = wsf;            // [512]
    float* msum    = wsf + 512;      // [1] (+pad to 768)
    float* centrs  = wsf + 768;      // [512*128] signed hyperboloid centroids
    float* out_graph = (float*)d_out;          // [512]
    float* out_node  = out_graph + NCENT;      // [N*512]

    init_ws_k<<<3, 256, 0, stream>>>(wsf);
    mask_sum_k<<<128, 256, 0, stream>>>(mask, nMask, msum);
    centroid_k<<<NCENT / 8, 256, 0, stream>>>(emb, centrs);
    dim3 grid((N + 63) / 64, NCENT / 32);
    dist_k<<<grid, 256, 0, stream>>>(node, centrs, out_node, colsum, N);
    finalize_k<<<2, 256, 0, stream>>>(colsum, msum, out_graph);
}